// GaussianMixtureUnif_80393197846622
// MI455X (gfx1250) — compile-verified
//
#include <hip/hip_runtime.h>
#include <math.h>

// ---------------------------------------------------------------------------
// GMM EM step for MI455X (gfx1250, wave32).
// GEMMs use V_WMMA_F32_16X16X4_F32 (native f32 matrix path, matches the f32
// reference precision).  gamma (N x C, f32, 256MB) lives in d_ws.
// LDS staging uses GLOBAL_LOAD_ASYNC_TO_LDS_B128 when the toolchain exposes
// the builtin (ASYNCcnt-tracked, no VGPR round-trip), else a sync fallback.
// ---------------------------------------------------------------------------

#define N_PTS 65536
#define C_CL  1024
#define D_DIM 128
#define HLOG2PI 0.91893853320467274178f
#define LOG_N  11.090354888959125f   // ln(65536)

typedef __attribute__((ext_vector_type(2))) float v2f;
typedef __attribute__((ext_vector_type(8))) float v8f;

#if __has_builtin(__builtin_amdgcn_global_load_async_to_lds_b128)
#define HAS_ASYNC_LDS 1
#else
#define HAS_ASYNC_LDS 0
#endif

#if HAS_ASYNC_LDS
typedef int v4i_vs __attribute__((vector_size(4 * sizeof(int))));
typedef __attribute__((address_space(1))) v4i_vs* g_v4i_p;
typedef __attribute__((address_space(3))) v4i_vs* l_v4i_p;
static __device__ __forceinline__ void async_g2l_b128(const void* g, void* l) {
  // (global src, lds dst, offset, cpol)
  __builtin_amdgcn_global_load_async_to_lds_b128(
      (g_v4i_p)(uintptr_t)g, (l_v4i_p)(uint32_t)(uintptr_t)l, 0, 0);
}
static __device__ __forceinline__ void wait_async0() {
#if __has_builtin(__builtin_amdgcn_s_wait_asynccnt)
  __builtin_amdgcn_s_wait_asynccnt(0);
#else
  asm volatile("s_wait_asynccnt 0x0" ::: "memory");
#endif
}
#endif

static __device__ __forceinline__ v8f wmma4(v2f a, v2f b, v8f c) {
  // D = A(16x4 f32) x B(4x16 f32) + C(16x16 f32)
  return __builtin_amdgcn_wmma_f32_16x16x4_f32(false, a, false, b,
                                               (short)0, c, false, false);
}

static __device__ __forceinline__ float wave_sum(float v) {
#pragma unroll
  for (int off = 16; off > 0; off >>= 1) v += __shfl_xor(v, off, 32);
  return v;
}
static __device__ __forceinline__ float half_max(float v) {
#pragma unroll
  for (int off = 8; off > 0; off >>= 1) v = fmaxf(v, __shfl_xor(v, off, 32));
  return v;
}
static __device__ __forceinline__ float half_sum(float v) {
#pragma unroll
  for (int off = 8; off > 0; off >>= 1) v += __shfl_xor(v, off, 32);
  return v;
}

// workspace layout (float offsets)
#define OFF_GBUF 0
#define OFF_X2   ((size_t)N_PTS * C_CL)
#define OFF_WCZ  (OFF_X2 + N_PTS)
#define OFF_M2   (OFF_WCZ + C_CL)
#define OFF_SC   (OFF_M2 + C_CL)
#define OFF_U    (OFF_SC + C_CL)
#define OFF_MUN  (OFF_U + (size_t)C_CL * D_DIM)
#define OFF_SCAL (OFF_MUN + (size_t)C_CL * D_DIM)
// scal: 0=accA (sum g*lg), 1=sigsum (sum g*D2), 2=sumY2, 3=sumXmY2,
//       4=term3 (sum S_c*(log S_c - log N)), 5=T1 (sum S_c*||mu_new||^2)

__global__ void k_zero(float* __restrict__ p, int n) {
  int i = blockIdx.x * blockDim.x + threadIdx.x;
  if (i < n) p[i] = 0.0f;
}

// one wave per row: x2[n] = ||X_n||^2 (rows < N), m2[c] = ||mu_c||^2 (rest)
__global__ void __launch_bounds__(256)
k_rownorm(const float* __restrict__ X, const float* __restrict__ mu,
          float* __restrict__ x2, float* __restrict__ m2) {
  int wave = threadIdx.x >> 5, lane = threadIdx.x & 31;
  int row = blockIdx.x * 8 + wave;
  const float* src; float* dst; int r;
  if (row < N_PTS) { src = X + (size_t)row * D_DIM; dst = x2; r = row; }
  else {
    r = row - N_PTS;
    if (r >= C_CL) return;
    src = mu + (size_t)r * D_DIM; dst = m2;
  }
  float s = 0.0f;
#pragma unroll
  for (int i = 0; i < 4; ++i) { float v = src[lane + i * 32]; s += v * v; }
  s = wave_sum(s);
  if (lane == 0) dst[r] = s;
}

// single block: Zw = logsumexp(w); wcz[c] = w[c] - Zw - D*(log(sigma)+HLOG2PI)
__global__ void __launch_bounds__(256)
k_wstats(const float* __restrict__ w, const float* __restrict__ sigma,
         float* __restrict__ wcz) {
  __shared__ float sh[8];
  __shared__ float sZ;
  int t = threadIdx.x, lane = t & 31, wave = t >> 5;
  float wl[4];
  float lm = -1e30f;
#pragma unroll
  for (int i = 0; i < 4; ++i) { wl[i] = w[t + i * 256]; lm = fmaxf(lm, wl[i]); }
#pragma unroll
  for (int off = 16; off > 0; off >>= 1) lm = fmaxf(lm, __shfl_xor(lm, off, 32));
  if (lane == 0) sh[wave] = lm;
  __syncthreads();
  float bm = sh[0];
#pragma unroll
  for (int i = 1; i < 8; ++i) bm = fmaxf(bm, sh[i]);
  __syncthreads();
  float ls = 0.0f;
#pragma unroll
  for (int i = 0; i < 4; ++i) ls += expf(wl[i] - bm);
  ls = wave_sum(ls);
  if (lane == 0) sh[wave] = ls;
  __syncthreads();
  if (t == 0) {
    float s = 0.0f;
    for (int i = 0; i < 8; ++i) s += sh[i];
    sZ = bm + logf(s);
  }
  __syncthreads();
  float sg = sigma[0];
  float lgn = (float)D_DIM * (logf(sg) + HLOG2PI);
  float Zw = sZ;
#pragma unroll
  for (int i = 0; i < 4; ++i) wcz[t + i * 256] = wl[i] - Zw - lgn;
}

// Main kernel: block = 16 rows x 1024 cols. 8 waves, wave w covers cols
// [w*128, w*128+128) as 8 WMMA tiles. P = X mu^T via f32 WMMA with mu staged
// through LDS. Then t = wcz_c - (x2+m2-2P)/(2s^2), row logsumexp across the
// block, gamma -> gbuf, S_c atomics, scalar accumulators.
__global__ void __launch_bounds__(256)
k_gamma(const float* __restrict__ X, const float* __restrict__ mu,
        const float* __restrict__ x2, const float* __restrict__ wcz,
        const float* __restrict__ m2, const float* __restrict__ sigma,
        float* __restrict__ gbuf, float* __restrict__ S_c,
        float* __restrict__ scal) {
  __shared__ float smu[C_CL * 8];     // 32 KB: mu[:, d0:d0+8]
  __shared__ float sred[16 * 8];
  __shared__ float srowmax[16];
  __shared__ float srowsum[16];

  const int t = threadIdx.x;
  const int lane = t & 31;
  const int wave = t >> 5;
  const int m = lane & 15;
  const int half = lane >> 4;
  const int kq = half * 2;
  const int n0 = blockIdx.x * 16;

  const float sg = sigma[0];
  const float inv2s2 = 1.0f / (2.0f * sg * sg);
  const float two_s2 = 2.0f * sg * sg;

  v8f acc[8] = {};

  for (int d0 = 0; d0 < D_DIM; d0 += 8) {
    // stage mu[:, d0:d0+8] -> smu[c*8 + dd]
#if HAS_ASYNC_LDS
#pragma unroll
    for (int i = 0; i < 8; ++i) {
      int idx4 = i * 256 + t;                // 2048 b128 slots
      int c = idx4 >> 1;
      int q = (idx4 & 1) * 4;
      async_g2l_b128(&mu[(size_t)c * D_DIM + d0 + q], &smu[c * 8 + q]);
    }
    wait_async0();
#else
#pragma unroll
    for (int i = 0; i < 16; ++i) {
      int idx2 = i * 256 + t;                // 4096 float2 slots
      int c = idx2 >> 2;
      int dp = (idx2 & 3) * 2;
      *(v2f*)&smu[c * 8 + dp] = *(const v2f*)&mu[(size_t)c * D_DIM + d0 + dp];
    }
#endif
    __syncthreads();
#pragma unroll
    for (int kk0 = 0; kk0 < 8; kk0 += 4) {
      int ka = kk0 + kq;
      v2f a = *(const v2f*)&X[(size_t)(n0 + m) * D_DIM + d0 + ka];
#pragma unroll
      for (int j = 0; j < 8; ++j) {
        int cl = wave * 128 + j * 16 + m;
        v2f b = *(const v2f*)&smu[cl * 8 + ka];
        acc[j] = wmma4(a, b, acc[j]);
      }
    }
    __syncthreads();
  }

  // per-element transform P -> t
  float xv[8];
#pragma unroll
  for (int v = 0; v < 8; ++v) xv[v] = x2[n0 + v + 8 * half];
  float wc[8], mm[8];
#pragma unroll
  for (int j = 0; j < 8; ++j) {
    int c = wave * 128 + j * 16 + m;
    wc[j] = wcz[c];
    mm[j] = m2[c];
  }
#pragma unroll
  for (int j = 0; j < 8; ++j)
#pragma unroll
    for (int v = 0; v < 8; ++v) {
      float P = acc[j][v];
      float d2 = xv[v] + mm[j] - 2.0f * P;
      acc[j][v] = wc[j] - inv2s2 * d2;
    }

  // row max across 128 cols of this wave, then across the 8 waves
  float lmax[8];
#pragma unroll
  for (int v = 0; v < 8; ++v) {
    float mx = acc[0][v];
#pragma unroll
    for (int j = 1; j < 8; ++j) mx = fmaxf(mx, acc[j][v]);
    lmax[v] = half_max(mx);
  }
  if (m == 0) {
#pragma unroll
    for (int v = 0; v < 8; ++v) sred[(v + 8 * half) * 8 + wave] = lmax[v];
  }
  __syncthreads();
  if (t < 16) {
    float mx = sred[t * 8];
    for (int wv = 1; wv < 8; ++wv) mx = fmaxf(mx, sred[t * 8 + wv]);
    srowmax[t] = mx;
  }
  __syncthreads();
  float gmax[8];
#pragma unroll
  for (int v = 0; v < 8; ++v) gmax[v] = srowmax[v + 8 * half];

  // row sum of exp(t - gmax)
  float lsum[8];
#pragma unroll
  for (int v = 0; v < 8; ++v) {
    float s = 0.0f;
#pragma unroll
    for (int j = 0; j < 8; ++j) s += expf(acc[j][v] - gmax[v]);
    lsum[v] = half_sum(s);
  }
  if (m == 0) {
#pragma unroll
    for (int v = 0; v < 8; ++v) sred[(v + 8 * half) * 8 + wave] = lsum[v];
  }
  __syncthreads();
  if (t < 16) {
    float s = 0.0f;
    for (int wv = 0; wv < 8; ++wv) s += sred[t * 8 + wv];
    srowsum[t] = s;
  }
  __syncthreads();
  float lrs[8];
#pragma unroll
  for (int v = 0; v < 8; ++v) lrs[v] = logf(srowsum[v + 8 * half]);

  // gamma, stores, accumulators
  float accA_l = 0.0f, sig_l = 0.0f;
#pragma unroll
  for (int j = 0; j < 8; ++j) {
    int c = wave * 128 + j * 16 + m;
    float csum = 0.0f;
#pragma unroll
    for (int v = 0; v < 8; ++v) {
      float tt = acc[j][v];
      float lg = (tt - gmax[v]) - lrs[v];     // log gamma
      float g = expf(lg);
      accA_l += g * lg;
      float d2 = (wc[j] - tt) * two_s2;       // recover D2
      sig_l += g * d2;
      csum += g;
      int row = n0 + v + 8 * half;
      gbuf[(size_t)row * C_CL + c] = g;
    }
    csum += __shfl_xor(csum, 16, 32);
    if (half == 0) unsafeAtomicAdd(&S_c[c], csum);
  }
  accA_l = wave_sum(accA_l);
  sig_l = wave_sum(sig_l);
  if (lane == 0) {
    unsafeAtomicAdd(&scal[0], accA_l);
    unsafeAtomicAdd(&scal[1], sig_l);
  }
}

// term3 = sum_c S_c * (log S_c - log N)
__global__ void __launch_bounds__(256)
k_term3(const float* __restrict__ S_c, float* __restrict__ scal) {
  int c = blockIdx.x * 256 + threadIdx.x;
  float s = S_c[c];
  float part = s * (logf(s) - LOG_N);
  part = wave_sum(part);
  if ((threadIdx.x & 31) == 0) unsafeAtomicAdd(&scal[4], part);
}

// U = gamma^T X (C x D), split-K with f32 atomic accumulation.
// One wave per block handles one c-tile and ALL 8 d-tiles: gamma is read
// exactly once (8 WMMAs amortize each gamma fragment load), X rows come from
// L2.  K-chunk of 2048, gridDim.y = 32 K-splits.
__global__ void __launch_bounds__(32)
k_ugemm(const float* __restrict__ gbuf, const float* __restrict__ X,
        float* __restrict__ U) {
  int lane = threadIdx.x;
  int m = lane & 15, half = lane >> 4, kq = half * 2;
  int c0 = blockIdx.x * 16;
  int k0 = blockIdx.y * 2048;
  v8f acc[8] = {};
  for (int k = k0; k < k0 + 2048; k += 4) {
    v2f a;
    a.x = gbuf[(size_t)(k + kq) * C_CL + c0 + m];
    a.y = gbuf[(size_t)(k + kq + 1) * C_CL + c0 + m];
    const float* xr = X + (size_t)(k + kq) * D_DIM + m;
#pragma unroll
    for (int j = 0; j < 8; ++j) {
      v2f b;
      b.x = xr[j * 16];
      b.y = xr[D_DIM + j * 16];
      acc[j] = wmma4(a, b, acc[j]);
    }
  }
#pragma unroll
  for (int j = 0; j < 8; ++j)
#pragma unroll
    for (int v = 0; v < 8; ++v)
      unsafeAtomicAdd(&U[(size_t)(c0 + v + 8 * half) * D_DIM + j * 16 + m],
                      acc[j][v]);
}

// mu_new = U / S_c ; T1 += S_c * ||mu_new_c||^2
__global__ void __launch_bounds__(128)
k_munew(const float* __restrict__ U, const float* __restrict__ S_c,
        float* __restrict__ mu_new, float* __restrict__ scal) {
  __shared__ float sp[4];
  int c = blockIdx.x, t = threadIdx.x;
  float s = S_c[c];
  float mn = U[(size_t)c * D_DIM + t] / s;
  mu_new[(size_t)c * D_DIM + t] = mn;
  float q = wave_sum(mn * mn);
  if ((t & 31) == 0) sp[t >> 5] = q;
  __syncthreads();
  if (t == 0) {
    float mn2 = sp[0] + sp[1] + sp[2] + sp[3];
    unsafeAtomicAdd(&scal[5], s * mn2);
  }
}

// Y = gamma mu_new (N x D). Block = 16 rows x 128 cols, LDS-staged A/B,
// fused sum||Y||^2 and sum||X-Y||^2.
__global__ void __launch_bounds__(256)
k_ygemm(const float* __restrict__ gbuf, const float* __restrict__ mu_new,
        const float* __restrict__ X, float* __restrict__ Y,
        float* __restrict__ scal) {
  __shared__ float sA[16 * 64];     // gamma[n0:n0+16, kc:kc+64]
  __shared__ float sB[64 * 128];    // mu_new[kc:kc+64, :]
  int t = threadIdx.x, lane = t & 31, wave = t >> 5;
  int m = lane & 15, half = lane >> 4, kq = half * 2;
  int n0 = blockIdx.x * 16;
  v8f acc = {};
  for (int kc = 0; kc < C_CL; kc += 64) {
#if HAS_ASYNC_LDS
    {
      int r = t >> 4, cp = (t & 15) * 4;     // 256 b128 slots for sA
      async_g2l_b128(&gbuf[(size_t)(n0 + r) * C_CL + kc + cp],
                     &sA[r * 64 + cp]);
    }
#pragma unroll
    for (int i = 0; i < 8; ++i) {
      int idx4 = i * 256 + t;                // 2048 b128 slots for sB
      int cr = idx4 >> 5, dp = (idx4 & 31) * 4;
      async_g2l_b128(&mu_new[(size_t)(kc + cr) * D_DIM + dp],
                     &sB[cr * 128 + dp]);
    }
    wait_async0();
#else
#pragma unroll
    for (int i = 0; i < 2; ++i) {
      int idx2 = i * 256 + t;
      int r = idx2 >> 5, cp = (idx2 & 31) * 2;
      *(v2f*)&sA[r * 64 + cp] =
          *(const v2f*)&gbuf[(size_t)(n0 + r) * C_CL + kc + cp];
    }
#pragma unroll
    for (int i = 0; i < 16; ++i) {
      int idx2 = i * 256 + t;
      int cr = idx2 >> 6, dp = (idx2 & 63) * 2;
      *(v2f*)&sB[cr * 128 + dp] =
          *(const v2f*)&mu_new[(size_t)(kc + cr) * D_DIM + dp];
    }
#endif
    __syncthreads();
#pragma unroll
    for (int kk = 0; kk < 64; kk += 4) {
      int ka = kk + kq;
      v2f a = *(const v2f*)&sA[m * 64 + ka];
      v2f b;
      b.x = sB[ka * 128 + wave * 16 + m];
      b.y = sB[(ka + 1) * 128 + wave * 16 + m];
      acc = wmma4(a, b, acc);
    }
    __syncthreads();
  }
  float y2 = 0.0f, xy2 = 0.0f;
  int d = wave * 16 + m;
#pragma unroll
  for (int v = 0; v < 8; ++v) {
    int n = n0 + v + 8 * half;
    float y = acc[v];
    Y[(size_t)n * D_DIM + d] = y;
    y2 += y * y;
    float x = X[(size_t)n * D_DIM + d];
    xy2 += (x - y) * (x - y);
  }
  y2 = wave_sum(y2);
  xy2 = wave_sum(xy2);
  if (lane == 0) {
    unsafeAtomicAdd(&scal[2], y2);
    unsafeAtomicAdd(&scal[3], xy2);
  }
}

__global__ void k_final(const float* __restrict__ scal,
                        const float* __restrict__ sigma,
                        float* __restrict__ out_scal) {
  float accA = scal[0], sigsum = scal[1], sumY2 = scal[2], sumXmY2 = scal[3];
  float term3 = scal[4], T1 = scal[5];
  float sn2 = sigsum / ((float)D_DIM * (float)N_PTS);   // sigma_new^2
  float inv2sn2 = 1.0f / (2.0f * sn2);
  float sg = sigma[0];
  float lgn = (float)D_DIM * (logf(sg) + HLOG2PI);      // uses OLD sigma
  float Cfe = inv2sn2 * (T1 - sumY2) + accA - term3 + (float)N_PTS * lgn;
  float FE = Cfe + inv2sn2 * sumXmY2;
  out_scal[0] = Cfe;
  out_scal[1] = FE;
}

extern "C" void kernel_launch(void* const* d_in, const int* in_sizes, int n_in,
                              void* d_out, int out_size, void* d_ws,
                              size_t ws_size, hipStream_t stream) {
  const float* X = (const float*)d_in[0];
  const float* mu = (const float*)d_in[1];
  const float* w = (const float*)d_in[2];
  const float* sigma = (const float*)d_in[3];
  float* out = (float*)d_out;
  float* ws = (float*)d_ws;

  float* gbuf = ws + OFF_GBUF;
  float* x2 = ws + OFF_X2;
  float* wcz = ws + OFF_WCZ;
  float* m2 = ws + OFF_M2;
  float* S_c = ws + OFF_SC;
  float* U = ws + OFF_U;
  float* mu_new = ws + OFF_MUN;
  float* scal = ws + OFF_SCAL;

  // zero S_c + U + mu_new + scal (contiguous accumulator region)
  int nz = C_CL + 2 * C_CL * D_DIM + 8;
  k_zero<<<(nz + 255) / 256, 256, 0, stream>>>(S_c, nz);
  k_rownorm<<<(N_PTS + C_CL) / 8, 256, 0, stream>>>(X, mu, x2, m2);
  k_wstats<<<1, 256, 0, stream>>>(w, sigma, wcz);
  k_gamma<<<N_PTS / 16, 256, 0, stream>>>(X, mu, x2, wcz, m2, sigma, gbuf, S_c,
                                          scal);
  k_term3<<<C_CL / 256, 256, 0, stream>>>(S_c, scal);
  k_ugemm<<<dim3(C_CL / 16, 32), 32, 0, stream>>>(gbuf, X, U);
  k_munew<<<C_CL, 128, 0, stream>>>(U, S_c, mu_new, scal);
  k_ygemm<<<N_PTS / 16, 256, 0, stream>>>(gbuf, mu_new, X, out, scal);
  k_final<<<1, 1, 0, stream>>>(scal, sigma, out + (size_t)N_PTS * D_DIM);
}